// MultifurcatingTreeModelLogReg_11673721110710
// MI455X (gfx1250) — compile-verified
//
#include <hip/hip_runtime.h>
#include <cmath>

typedef float v2f __attribute__((ext_vector_type(2)));
typedef float v8f __attribute__((ext_vector_type(8)));

constexpr int kNodes     = 1365;
constexpr int kFirstLeaf = 341;   // first level-5 (leaf) node id
constexpr int kFirstL4   = 85;    // first level-4 node id
constexpr int kLeaves    = 1024;
constexpr int kF         = 512;
constexpr int kC         = 64;
constexpr int kXsStride  = 132;   // 128 + 4 pad -> bank-conflict-free column reads
constexpr int kSlots     = 16;    // l1 partial slots per node

// workspace layout (floats)
constexpr int kLogitsA  = 64 * 4 * 16 * kC;   // [tile][ksplit][row][col] = 262144
constexpr int kLogitsB  = kLeaves * kC;        // 65536 (each for B5 and B4)
constexpr int kL1Elems  = kNodes * kSlots;     // 21840
// total floats = 262144 + 2*65536 + 21840 + 16 = 415072  (~1.66 MB)

// ---------------------------------------------------------------------------
__global__ void zero_ws_kernel(float* ws, int n) {
  int i = blockIdx.x * blockDim.x + threadIdx.x;
  if (i < n) ws[i] = 0.0f;
}

// ---------------------------------------------------------------------------
// Kernel A: WMMA over ancestor levels 0..3.
// grid = (4 ksplits, 64 leaf tiles), block = 128 threads (4 waves).
// Each wave owns a 16-column slice; K-slice of 128 per workgroup.
// Fuses L1(|delta|) and root sum-of-squares partials.
// ---------------------------------------------------------------------------
__global__ __launch_bounds__(128)
void wmma_levels_kernel(const float* __restrict__ x,
                        const float* __restrict__ deltas,
                        float* __restrict__ logitsA,
                        float* __restrict__ l1part,
                        float* __restrict__ sqpart)
{
  __shared__ float xs[16 * kXsStride];
  const int s   = blockIdx.x;          // ksplit 0..3
  const int t   = blockIdx.y;          // leaf tile 0..63
  const int tid = threadIdx.x;
  const int kbase = s * 128;

  // stage x[16t..16t+15][kbase..kbase+127] into LDS (float4)
  const float* xtile = x + (size_t)(16 * t) * kF + kbase;
  for (int idx = tid; idx < 16 * 32; idx += 128) {
    int row = idx >> 5, fq = idx & 31;
    float4 v = ((const float4*)(xtile + (size_t)row * kF))[fq];
    float* dst = &xs[row * kXsStride + fq * 4];
    dst[0] = v.x; dst[1] = v.y; dst[2] = v.z; dst[3] = v.w;
  }
  __syncthreads();

  const int wave = tid >> 5;           // 0..3 -> column tile
  const int lane = tid & 31;
  const int col  = wave * 16 + (lane & 15);
  const int row  = lane & 15;
  const int koff = (lane >> 4) * 2;    // f32 A/B fragment layout

  int nodes[4], own[4];
  nodes[0] = 0;             own[0] = (t == 0);
  nodes[1] = 1 + (t >> 4);  own[1] = ((t & 15) == 0);
  nodes[2] = 5 + (t >> 2);  own[2] = ((t & 3) == 0);
  nodes[3] = 21 + t;        own[3] = 1;

  v8f acc = {0.f, 0.f, 0.f, 0.f, 0.f, 0.f, 0.f, 0.f};

  for (int lev = 0; lev < 4; ++lev) {
    const float* db = deltas + (size_t)nodes[lev] * (kF * kC) + (size_t)kbase * kC;
    float myl1 = 0.f, mysq = 0.f;
    #pragma unroll 4
    for (int k = 0; k < 128; k += 4) {
      v2f a, b;
      a[0] = xs[row * kXsStride + k + koff];
      a[1] = xs[row * kXsStride + k + koff + 1];
      b[0] = db[(size_t)(k + koff) * kC + col];
      b[1] = db[(size_t)(k + koff + 1) * kC + col];
      acc = __builtin_amdgcn_wmma_f32_16x16x4_f32(
                false, a, false, b, (short)0, acc, false, false);
      myl1 += __builtin_fabsf(b[0]) + __builtin_fabsf(b[1]);
      mysq += b[0] * b[0] + b[1] * b[1];
    }
    if (own[lev]) {  // one tile owns each node; each ksplit covers distinct K
      float r = myl1;
      for (int off = 16; off > 0; off >>= 1) r += __shfl_xor(r, off, 32);
      if (lane == 0) l1part[nodes[lev] * kSlots + s * 4 + wave] = r;
    }
    if (lev == 0 && t == 0) {
      float r = mysq;
      for (int off = 16; off > 0; off >>= 1) r += __shfl_xor(r, off, 32);
      if (lane == 0) sqpart[s * 4 + wave] = r;
    }
  }

  // C/D layout: VGPR r -> M=r (lanes 0-15) / M=r+8 (lanes 16-31)
  const int rbase = (lane >> 4) * 8;
  float* outA = logitsA + (size_t)((t * 4 + s) * 16) * kC;
  for (int r = 0; r < 8; ++r) outA[(size_t)(rbase + r) * kC + col] = acc[r];
}

// ---------------------------------------------------------------------------
// Kernel B: levels 4 and 5 — one workgroup per node (1280 WGs, 256 threads).
// Thread layout: 16 col-groups (float4 => global_load_b128) x 16 f-chunks.
// Fuses per-node L1; per-leaf logits reduced across f-chunks in LDS.
// ---------------------------------------------------------------------------
__global__ __launch_bounds__(256)
void leaf_levels_kernel(const float* __restrict__ x,
                        const float* __restrict__ deltas,
                        float* __restrict__ logitsB5,
                        float* __restrict__ logitsB4,
                        float* __restrict__ l1part)
{
  __shared__ float xs[4][kF];          // up to 4 leaf rows of x
  __shared__ float redq[16][kC];       // cross-f-chunk reduction

  const int node = kFirstL4 + blockIdx.x;       // 85..1364
  const int tid  = threadIdx.x;
  const int cg   = (tid & 15) * 4;     // column group (4 cols)
  const int q    = tid >> 4;           // f-chunk 0..15 (32 f each)
  const int lane = tid & 31;
  const int wave = tid >> 5;           // 0..7
  const int fbase = q * 32;

  const bool isL4 = (node < kFirstLeaf);
  const int nleaf = isL4 ? 4 : 1;
  const int leaf0 = isL4 ? 4 * (node - kFirstL4) : (node - kFirstLeaf);

  // stage x rows
  for (int idx = tid; idx < nleaf * (kF / 4); idx += 256) {
    int row = idx >> 7, fq = idx & 127;
    float4 v = ((const float4*)(x + (size_t)(leaf0 + row) * kF))[fq];
    float* dst = &xs[row][fq * 4];
    dst[0] = v.x; dst[1] = v.y; dst[2] = v.z; dst[3] = v.w;
  }
  __syncthreads();

  const float* dp = deltas + ((size_t)node * kF + fbase) * kC + cg;
  float myl1 = 0.f;

  if (isL4) {
    float acc[4][4];
    #pragma unroll
    for (int j = 0; j < 4; ++j)
      for (int v = 0; v < 4; ++v) acc[j][v] = 0.f;
    #pragma unroll 2
    for (int ff = 0; ff < 32; ++ff) {
      float4 d = *(const float4*)(dp + (size_t)ff * kC);
      myl1 += __builtin_fabsf(d.x) + __builtin_fabsf(d.y) +
              __builtin_fabsf(d.z) + __builtin_fabsf(d.w);
      #pragma unroll
      for (int j = 0; j < 4; ++j) {
        float xv = xs[j][fbase + ff];
        acc[j][0] += xv * d.x; acc[j][1] += xv * d.y;
        acc[j][2] += xv * d.z; acc[j][3] += xv * d.w;
      }
    }
    // reduce across 16 f-chunks per leaf, write logitsB4
    #pragma unroll
    for (int j = 0; j < 4; ++j) {
      redq[q][cg + 0] = acc[j][0]; redq[q][cg + 1] = acc[j][1];
      redq[q][cg + 2] = acc[j][2]; redq[q][cg + 3] = acc[j][3];
      __syncthreads();
      if (tid < kC) {
        float ssum = 0.f;
        for (int qq = 0; qq < 16; ++qq) ssum += redq[qq][tid];
        logitsB4[(size_t)(leaf0 + j) * kC + tid] = ssum;
      }
      __syncthreads();
    }
  } else {
    float a0 = 0.f, a1 = 0.f, a2 = 0.f, a3 = 0.f;
    #pragma unroll 2
    for (int ff = 0; ff < 32; ++ff) {
      float4 d = *(const float4*)(dp + (size_t)ff * kC);
      myl1 += __builtin_fabsf(d.x) + __builtin_fabsf(d.y) +
              __builtin_fabsf(d.z) + __builtin_fabsf(d.w);
      float xv = xs[0][fbase + ff];
      a0 += xv * d.x; a1 += xv * d.y; a2 += xv * d.z; a3 += xv * d.w;
    }
    redq[q][cg + 0] = a0; redq[q][cg + 1] = a1;
    redq[q][cg + 2] = a2; redq[q][cg + 3] = a3;
    __syncthreads();
    if (tid < kC) {
      float ssum = 0.f;
      for (int qq = 0; qq < 16; ++qq) ssum += redq[qq][tid];
      logitsB5[(size_t)leaf0 * kC + tid] = ssum;
    }
    __syncthreads();
  }

  // per-node L1: 8 wave partials (slots 0..7; 8..15 stay zero)
  for (int off = 16; off > 0; off >>= 1) myl1 += __shfl_xor(myl1, off, 32);
  if (lane == 0) l1part[node * kSlots + wave] = myl1;
}

// ---------------------------------------------------------------------------
// Kernel C: combine partial logits + per-row softmax -> out
// grid = 64 tiles, block = 256 (16 rows x 16 lanes, 4 cols each)
// ---------------------------------------------------------------------------
__global__ __launch_bounds__(256)
void combine_softmax_kernel(const float* __restrict__ logitsA,
                            const float* __restrict__ logitsB5,
                            const float* __restrict__ logitsB4,
                            float* __restrict__ out)
{
  __shared__ float red[16 * 16];
  const int t   = blockIdx.x;
  const int tid = threadIdx.x;
  const int r   = tid >> 4;
  const int ln  = tid & 15;
  const int leaf = 16 * t + r;

  float v[4];
  #pragma unroll
  for (int vv = 0; vv < 4; ++vv) {
    const int c = ln + 16 * vv;
    float s = logitsB5[(size_t)leaf * kC + c] + logitsB4[(size_t)leaf * kC + c];
    #pragma unroll
    for (int sp = 0; sp < 4; ++sp)
      s += logitsA[(size_t)((t * 4 + sp) * 16 + r) * kC + c];
    v[vv] = s;
  }

  red[r * 16 + ln] = fmaxf(fmaxf(v[0], v[1]), fmaxf(v[2], v[3]));
  __syncthreads();
  float rmax = red[r * 16];
  for (int i = 1; i < 16; ++i) rmax = fmaxf(rmax, red[r * 16 + i]);
  __syncthreads();
  float e0 = expf(v[0] - rmax), e1 = expf(v[1] - rmax);
  float e2 = expf(v[2] - rmax), e3 = expf(v[3] - rmax);
  red[r * 16 + ln] = e0 + e1 + e2 + e3;
  __syncthreads();
  float ssum = 0.f;
  for (int i = 0; i < 16; ++i) ssum += red[r * 16 + i];
  const float inv = 1.f / ssum;
  float* o = out + (size_t)leaf * kC;
  o[ln] = e0 * inv; o[ln + 16] = e1 * inv;
  o[ln + 32] = e2 * inv; o[ln + 48] = e3 * inv;
}

// ---------------------------------------------------------------------------
// Kernel D: delta_tensor = sum(sq(root)) + sum_{n>0} l1[n]/max(|dh|,1e-7)
// ---------------------------------------------------------------------------
__global__ void reg_kernel(const float* __restrict__ heights,
                           const int*   __restrict__ parent,
                           const float* __restrict__ l1part,
                           const float* __restrict__ sqpart,
                           float* __restrict__ out_scalar)
{
  __shared__ float sred[256];
  const int tid = threadIdx.x;
  float p = 0.f;
  for (int n = tid; n < kNodes; n += 256) {
    float s = 0.f;
    for (int k = 0; k < kSlots; ++k) s += l1part[n * kSlots + k];
    if (n > 0) {
      float d = fmaxf(fabsf(heights[n] - heights[parent[n]]), 1e-7f);
      p += s / d;
    }
  }
  sred[tid] = p;
  __syncthreads();
  for (int off = 128; off > 0; off >>= 1) {
    if (tid < off) sred[tid] += sred[tid + off];
    __syncthreads();
  }
  if (tid == 0) {
    float sq = 0.f;
    for (int k = 0; k < kSlots; ++k) sq += sqpart[k];
    out_scalar[0] = sq + sred[0];
  }
}

// ---------------------------------------------------------------------------
extern "C" void kernel_launch(void* const* d_in, const int* in_sizes, int n_in,
                              void* d_out, int out_size, void* d_ws, size_t ws_size,
                              hipStream_t stream) {
  const float* x       = (const float*)d_in[0];
  const float* deltas  = (const float*)d_in[1];
  const float* heights = (const float*)d_in[2];
  const int*   parent  = (const int*)  d_in[4];
  float* out = (float*)d_out;

  float* logitsA  = (float*)d_ws;
  float* logitsB5 = logitsA + kLogitsA;
  float* logitsB4 = logitsB5 + kLogitsB;
  float* l1part   = logitsB4 + kLogitsB;
  float* sqpart   = l1part + kL1Elems;

  const int nzero = kL1Elems + kSlots;
  zero_ws_kernel<<<(nzero + 255) / 256, 256, 0, stream>>>(l1part, nzero);
  wmma_levels_kernel<<<dim3(4, 64), 128, 0, stream>>>(x, deltas, logitsA, l1part, sqpart);
  leaf_levels_kernel<<<kNodes - kFirstL4, 256, 0, stream>>>(x, deltas, logitsB5, logitsB4, l1part);
  combine_softmax_kernel<<<64, 256, 0, stream>>>(logitsA, logitsB5, logitsB4, out);
  reg_kernel<<<1, 256, 0, stream>>>(heights, parent, l1part, sqpart, out + kLeaves * kC);
}